// NeuralODE_36670430773281
// MI455X (gfx1250) — compile-verified
//
#include <hip/hip_runtime.h>

typedef __attribute__((ext_vector_type(2))) float v2f;
typedef __attribute__((ext_vector_type(8))) float v8f;

#define D    128
#define STR  130          // padded LDS row stride (floats): avoids 16-way bank conflicts
#define YLEN (D + D * D)  // 16512
#define NT   512          // 16 wave32 waves
#define TPW  4            // 64 output tiles / 16 waves

__launch_bounds__(NT, 1)
__global__ void NeuralODE_tsit5_persistent(const float* __restrict__ ts,
                                           const float* __restrict__ y0,
                                           const float* __restrict__ Bm,
                                           float* __restrict__ out, int T)
{
    __shared__ float Wl[D * STR];
    __shared__ float Bl[D * STR];
    __shared__ float Sl[D * STR];
    __shared__ float xl[D];
    __shared__ float xs[D];
    __shared__ float kx[6][D];
    __shared__ float kxp[4 * D];

    const int tid  = threadIdx.x;
    const int lane = tid & 31;
    const int wv   = tid >> 5;
    const int nn   = lane & 15;  // row/col index within 16-wide tile
    const int hf   = lane >> 4;  // half-wave selector

    // ---- init: stage y0 and B into LDS; emit output row 0 ----
    for (int i = tid; i < D * D; i += NT) {
        int r = i >> 7, c = i & 127;
        float w = y0[D + i];
        Wl[r * STR + c] = w;
        Bl[r * STR + c] = Bm[i];
        out[(size_t)D + i] = w;
    }
    if (tid < D) { xl[tid] = y0[tid]; out[tid] = y0[tid]; }
    __syncthreads();

    constexpr float AC[6][5] = {
        {0.f, 0.f, 0.f, 0.f, 0.f},
        {0.161f, 0.f, 0.f, 0.f, 0.f},
        {-0.008480655492356989f, 0.335480655492357f, 0.f, 0.f, 0.f},
        {2.8971530571054935f, -6.359448489975075f, 4.3622954328695815f, 0.f, 0.f},
        {5.325864828439257f, -11.748883564062828f, 7.4955393428898365f, -0.09249506636175525f, 0.f},
        {5.86145544294642f, -12.92096931784711f, 8.159367898576159f, -0.071584973281401f, -0.028269050394068383f}};
    constexpr float BC[6] = {0.09646076681806523f, 0.01f, 0.4798896504144996f,
                             1.379008574103742f, -3.290069515436081f, 2.324710524099774f};

    float kreg[5][TPW][8];   // kW_1..kW_5 tiles, register-resident
    float ksum[TPW][8];      // running sum of b_i * kW_i

    for (int n = 0; n < T - 1; ++n) {
        const float dt = ts[n + 1] - ts[n];

        #pragma unroll
        for (int t = 0; t < TPW; ++t)
            #pragma unroll
            for (int r = 0; r < 8; ++r) ksum[t][r] = 0.f;

        #pragma unroll
        for (int i = 0; i < 6; ++i) {
            // ---- build stage inputs S_i = W + dt*sum_j a_ij kW_j ; xs_i likewise ----
            if (i > 0) {
                #pragma unroll
                for (int t = 0; t < TPW; ++t) {
                    const int g = wv * TPW + t;
                    const int tm = g >> 3, tn = g & 7;
                    #pragma unroll
                    for (int r = 0; r < 8; ++r) {
                        const int row = tm * 16 + r + 8 * hf;
                        const int col = tn * 16 + nn;
                        float acc = 0.f;
                        #pragma unroll
                        for (int j = 0; j < i; ++j) acc += AC[i][j] * kreg[j][t][r];
                        Sl[row * STR + col] = Wl[row * STR + col] + dt * acc;
                    }
                }
                if (tid < D) {
                    float acc = 0.f;
                    #pragma unroll
                    for (int j = 0; j < i; ++j) acc += AC[i][j] * kx[j][tid];
                    xs[tid] = xl[tid] + dt * acc;
                }
            }
            __syncthreads();

            const float* Sp = (i == 0) ? Wl : Sl;
            const float* xp = (i == 0) ? xl : xs;

            // ---- kW_i = S_i @ B via chained V_WMMA_F32_16X16X4_F32 (K = 128) ----
            #pragma unroll
            for (int t = 0; t < TPW; ++t) {
                const int g = wv * TPW + t;
                const int tm = g >> 3, tn = g & 7;
                v8f acc = {};
                const float* arow = Sp + (tm * 16 + nn) * STR + 2 * hf;   // A: 16x4 slice
                const float* bcol = Bl + (2 * hf) * STR + tn * 16 + nn;   // B: 4x16 slice
                #pragma unroll 8
                for (int kk = 0; kk < D; kk += 4) {
                    v2f a, b;
                    a.x = arow[kk];
                    a.y = arow[kk + 1];
                    b.x = bcol[kk * STR];
                    b.y = bcol[kk * STR + STR];
                    acc = __builtin_amdgcn_wmma_f32_16x16x4_f32(
                        false, a, false, b, (short)0, acc, false, false);
                }
                #pragma unroll
                for (int r = 0; r < 8; ++r) {
                    ksum[t][r] += BC[i] * acc[r];
                    if (i < 5) kreg[i][t][r] = acc[r];
                }
            }

            // ---- kx_i = S_i @ xs_i : split-row matvec, 4 partials per row ----
            {
                const int row  = tid & (D - 1);
                const int part = tid >> 7;  // 0..3
                const float* srow = Sp + row * STR + part * 32;
                const float* xv   = xp + part * 32;
                float s = 0.f;
                #pragma unroll 8
                for (int c = 0; c < 32; ++c) s += srow[c] * xv[c];
                kxp[part * D + row] = s;
            }
            __syncthreads();
            if (tid < D)
                kx[i][tid] = kxp[tid] + kxp[D + tid] + kxp[2 * D + tid] + kxp[3 * D + tid];
            __syncthreads();
        }

        // ---- y_{n+1}: W += dt * sum b_i kW_i ; x += dt * sum b_i kx_i ; stream out ----
        float* orow = out + (size_t)(n + 1) * YLEN;
        #pragma unroll
        for (int t = 0; t < TPW; ++t) {
            const int g = wv * TPW + t;
            const int tm = g >> 3, tn = g & 7;
            #pragma unroll
            for (int r = 0; r < 8; ++r) {
                const int row = tm * 16 + r + 8 * hf;
                const int col = tn * 16 + nn;
                float w = Wl[row * STR + col] + dt * ksum[t][r];
                Wl[row * STR + col] = w;
                orow[D + row * D + col] = w;
            }
        }
        if (tid < D) {
            float acc = 0.f;
            #pragma unroll
            for (int i = 0; i < 6; ++i) acc += BC[i] * kx[i][tid];
            float xn = xl[tid] + dt * acc;
            xl[tid] = xn;
            orow[tid] = xn;
        }
        __syncthreads();
    }
}

extern "C" void kernel_launch(void* const* d_in, const int* in_sizes, int n_in,
                              void* d_out, int out_size, void* d_ws, size_t ws_size,
                              hipStream_t stream) {
    const float* ts = (const float*)d_in[0];
    const float* y0 = (const float*)d_in[1];
    const float* Bm = (const float*)d_in[2];
    float* out      = (float*)d_out;
    const int T     = in_sizes[0];  // 4096
    NeuralODE_tsit5_persistent<<<1, NT, 0, stream>>>(ts, y0, Bm, out, T);
}